// VectorQuantizer_49615462203698
// MI455X (gfx1250) — compile-verified
//
#include <hip/hip_runtime.h>

typedef __attribute__((ext_vector_type(2))) float v2f;
typedef __attribute__((ext_vector_type(4))) float v4f;
typedef __attribute__((ext_vector_type(8))) float v8f;
typedef __attribute__((ext_vector_type(4))) unsigned int u32x4;
typedef __attribute__((ext_vector_type(8))) int i32x8;
typedef __attribute__((ext_vector_type(4))) int i32x4;

#define NUM_K   4096
#define DIM     256
#define BATCH   32768
#define TILE_M  128            // z rows per workgroup (8 waves x 16)
#define TILE_N  64             // codebook rows per LDS stage
#define N_TILES (NUM_K / TILE_N)
#define LDS_PITCH  260         // 256 + 4 pad floats -> conflict-free transposed b64 reads
#define LDS_FLOATS (TILE_N * LDS_PITCH)          // one buffer, floats
#define LDS_BYTES  (2 * LDS_FLOATS * 4)          // double-buffered: 133120 B -> 2 WGs/WGP

// ---------------------------------------------------------------- w_sq[k]
__global__ void vq_wsq_kernel(const float* __restrict__ cb, float* __restrict__ wsq) {
    const int wave = threadIdx.x >> 5;
    const int lane = threadIdx.x & 31;
    const int row  = blockIdx.x * 8 + wave;
    const float* r = cb + (size_t)row * DIM;
    float s = 0.f;
#pragma unroll
    for (int j = 0; j < 8; ++j) { float v = r[lane + 32 * j]; s += v * v; }
#pragma unroll
    for (int m = 16; m >= 1; m >>= 1) s += __shfl_xor(s, m, 32);
    if (lane == 0) wsq[row] = s;
}

// TDM: stage one 64x256 f32 codebook tile into LDS with a 4-DWORD pad after
// every 256-DWORD row (reproduces the 260-float bank-conflict-free pitch).
// D# per CDNA5 ISA 8.3/8.4: group0 = {count=1, lds_addr, global_addr, type=2},
// group1 = {data_size=4B, pad_enable, pad_interval=256dw, pad_amount=4dw,
//           tensor_dim0=256, tensor_dim1=64, tile_dim0=256, tile_dim1=64,
//           tensor_dim0_stride=256}; trailing groups zero (2D tensor).
__device__ __forceinline__ void tdm_stage_tile(const float* gsrc, unsigned lds_off) {
    unsigned long long ga = (unsigned long long)gsrc;
    u32x4 g0 = { 1u,                                    // count=1 (valid)
                 lds_off,                               // lds_addr (bytes)
                 (unsigned)(ga & 0xFFFFFFFFu),          // global_addr[31:0]
                 (unsigned)((ga >> 32) & 0x01FFFFFFu)   // global_addr[56:32]
                     | 0x80000000u };                   // type=2 ("image")
    i32x8 g1 = { (int)0x07D20000,   // data_size=2(4B) | pad_en | pad_int=7 | pad_amt=3
                 (int)0x01000000,   // tensor_dim0 = 256 (low16 in [31:16])
                 (int)0x00400000,   // tensor_dim1 = 64  (low16 in [31:16])
                 (int)0x01000000,   // tile_dim0   = 256 (in [31:16])
                 64,                // tile_dim1 = 64, tile_dim2 = 0
                 256,               // tensor_dim0_stride = 256
                 0, 0 };
    i32x4 gz4 = { 0, 0, 0, 0 };
    i32x8 gz8 = { 0, 0, 0, 0, 0, 0, 0, 0 };
    __builtin_amdgcn_tensor_load_to_lds(g0, g1, gz4, gz4, gz8, 0);
}

// ------------------------------------------------- fused GEMM + argmin (WMMA f32)
__global__ __launch_bounds__(256) void vq_argmin_kernel(
    const float* __restrict__ z, const float* __restrict__ cb,
    const float* __restrict__ wsq, int* __restrict__ idx_out) {
    extern __shared__ float lds[];          // 2 x [TILE_N][LDS_PITCH]

    const int tid  = threadIdx.x;
    const int wave = tid >> 5;
    const int lane = tid & 31;
    const int nl   = lane & 15;             // N within 16x16 tile / M-row for A loads
    const int hi   = lane >> 4;             // half-wave selector
    const int m_wave = blockIdx.x * TILE_M + wave * 16;

    // A fragments: lane carries z[m_wave+nl, 4s + 2*hi + {0,1}], s=0..63.
    // Entire 16x256 A tile register-resident: 64 x v2f = 128 VGPRs.
    v2f a[64];
    const float* zr = z + (size_t)(m_wave + nl) * DIM + 2 * hi;
#pragma unroll
    for (int s = 0; s < 64; ++s)
        a[s] = *reinterpret_cast<const v2f*>(zr + 4 * s);

    float best[8];
    int   bidx[8];
#pragma unroll
    for (int v = 0; v < 8; ++v) { best[v] = 3.4e38f; bidx[v] = 0; }

    // ---- prologue: per-lane async-stage tile 0 into buffer 0 (ASYNCcnt path) ----
    {
        const int c4   = tid & 63;      // float4 column
        const int row0 = tid >> 6;      // rows row0 + 4*i
        unsigned ldsoff = (unsigned)(size_t)&lds[row0 * LDS_PITCH + c4 * 4];
        unsigned long long g =
            (unsigned long long)(cb + (size_t)row0 * DIM + c4 * 4);
#pragma unroll
        for (int i = 0; i < TILE_N / 4; ++i) {
            asm volatile("global_load_async_to_lds_b128 %0, %1, off"
                         :: "v"(ldsoff), "v"(g) : "memory");
            ldsoff += 4 * LDS_PITCH * 4;   // 4 padded rows
            g      += 4 * DIM * 4;         // 4 tensor rows
        }
    }
    asm volatile("s_wait_asynccnt 0x0" ::: "memory");
    __syncthreads();

    for (int t = 0; t < N_TILES; ++t) {
        const int cur = t & 1;
        // ---- TDM-prefetch next tile into the other buffer (TENSORcnt path) ----
        if (wave == 0 && t + 1 < N_TILES) {
            tdm_stage_tile(cb + (size_t)(t + 1) * TILE_N * DIM,
                           (unsigned)(size_t)&lds[(cur ^ 1) * LDS_FLOATS]);
        }

        const int    n0   = t * TILE_N;
        const float* base = &lds[cur * LDS_FLOATS];
#pragma unroll 1
        for (int nt = 0; nt < TILE_N / 16; ++nt) {
            // B fragment: lane carries codebook[n0+nt*16+nl, 4s + 2*hi + {0,1}]
            const unsigned bpo = (unsigned)(size_t)
                (base + (nt * 16 + nl) * LDS_PITCH + 2 * hi);

            // Manually pipelined: asm ds_load_b64 chunks of 8 k-steps,
            // double-buffered; explicit s_wait_dscnt with pass-through ties
            // so WMMAs can't be hoisted above the wait.
            v2f bb[2][8];
#pragma unroll
            for (int i = 0; i < 8; ++i)
                asm volatile("ds_load_b64 %0, %1 offset:%2"
                             : "=v"(bb[0][i]) : "v"(bpo), "n"(16 * i));
            v8f c = {};
#pragma unroll
            for (int g8 = 0; g8 < 8; ++g8) {
                const int curb = g8 & 1;
                if (g8 < 7) {
#pragma unroll
                    for (int i = 0; i < 8; ++i)
                        asm volatile("ds_load_b64 %0, %1 offset:%2"
                                     : "=v"(bb[curb ^ 1][i])
                                     : "v"(bpo), "n"(128 * (g8 + 1) + 16 * i));
                    asm volatile("s_wait_dscnt 0x8"
                                 : "+v"(bb[curb][0]), "+v"(bb[curb][1]),
                                   "+v"(bb[curb][2]), "+v"(bb[curb][3]),
                                   "+v"(bb[curb][4]), "+v"(bb[curb][5]),
                                   "+v"(bb[curb][6]), "+v"(bb[curb][7]));
                } else {
                    asm volatile("s_wait_dscnt 0x0"
                                 : "+v"(bb[curb][0]), "+v"(bb[curb][1]),
                                   "+v"(bb[curb][2]), "+v"(bb[curb][3]),
                                   "+v"(bb[curb][4]), "+v"(bb[curb][5]),
                                   "+v"(bb[curb][6]), "+v"(bb[curb][7]));
                }
#pragma unroll
                for (int i = 0; i < 8; ++i)
                    c = __builtin_amdgcn_wmma_f32_16x16x4_f32(
                            false, a[8 * g8 + i], false, bb[curb][i],
                            (short)0, c, false, false);
            }
            const int   n  = n0 + nt * 16 + nl;
            const float wq = wsq[n];
            // d2 (minus per-row const ||z||^2): lane holds rows v + 8*hi, col n
#pragma unroll
            for (int v = 0; v < 8; ++v) {
                float d = wq - 2.0f * c[v];
                if (d < best[v]) { best[v] = d; bidx[v] = n; }
            }
        }
        if (wave == 0) __builtin_amdgcn_s_wait_tensorcnt(0);
        __syncthreads();
    }

    // min-reduce across the 16 lanes of each half (xor masks < 16 stay in-half)
#pragma unroll
    for (int v = 0; v < 8; ++v) {
        float bv = best[v];
        int   bi = bidx[v];
#pragma unroll
        for (int m = 8; m >= 1; m >>= 1) {
            float ov = __shfl_xor(bv, m, 32);
            int   oi = __shfl_xor(bi, m, 32);
            if (ov < bv || (ov == bv && oi < bi)) { bv = ov; bi = oi; }
        }
        if (nl == 0) idx_out[m_wave + hi * 8 + v] = bi;
    }
}

// ---------------------------------- gather quantized rows + per-block SSE partial
__global__ void vq_gather_kernel(const float* __restrict__ z,
                                 const float* __restrict__ cb,
                                 const int* __restrict__ idx,
                                 float* __restrict__ out,
                                 float* __restrict__ partial) {
    __shared__ float red[8];
    const int wave = threadIdx.x >> 5;
    const int lane = threadIdx.x & 31;
    const int row  = blockIdx.x * 8 + wave;
    const int k    = idx[row];
    const v4f* zr  = reinterpret_cast<const v4f*>(z  + (size_t)row * DIM);
    const v4f* qr  = reinterpret_cast<const v4f*>(cb + (size_t)k   * DIM);
    v4f*       orw = reinterpret_cast<v4f*>(out + (size_t)row * DIM);
    float s = 0.f;
#pragma unroll
    for (int i = 0; i < 2; ++i) {
        int c4 = lane + 32 * i;
        v4f q  = qr[c4];
        v4f zv = zr[c4];
        orw[c4] = q;
        v4f d = zv - q;
        s += d.x * d.x + d.y * d.y + d.z * d.z + d.w * d.w;
    }
#pragma unroll
    for (int m = 16; m >= 1; m >>= 1) s += __shfl_xor(s, m, 32);
    if (lane == 0) red[wave] = s;
    __syncthreads();
    if (threadIdx.x == 0) {
        float t = 0.f;
#pragma unroll
        for (int w = 0; w < 8; ++w) t += red[w];
        partial[blockIdx.x] = t;   // deterministic: no float atomics
    }
}

// ------------------------------------------------ fixed-order loss finalization
__global__ void vq_final_kernel(const float* __restrict__ partial,
                                float* __restrict__ loss_out) {
    float t = 0.f;
    for (int i = 0; i < BATCH / 8; ++i) t += partial[i];
    // vq_loss = (1 + COMMITMENT_COST) * mean((z - q)^2)
    *loss_out = 1.25f * t / (float)(BATCH * DIM);
}

extern "C" void kernel_launch(void* const* d_in, const int* in_sizes, int n_in,
                              void* d_out, int out_size, void* d_ws, size_t ws_size,
                              hipStream_t stream) {
    const float* z  = (const float*)d_in[0];   // [32768, 256] f32
    const float* cb = (const float*)d_in[1];   // [4096, 256]  f32
    float* out = (float*)d_out;                // [32768*256] quantized_st + [1] loss

    char*  ws      = (char*)d_ws;
    int*   idx     = (int*)ws;                                     // 128 KB
    float* wsq     = (float*)(ws + (size_t)BATCH * 4);             // 16 KB
    float* partial = (float*)(ws + (size_t)BATCH * 4 + NUM_K * 4); // 16 KB

    vq_wsq_kernel   <<<NUM_K / 8, 256, 0, stream>>>(cb, wsq);
    vq_argmin_kernel<<<BATCH / TILE_M, 256, LDS_BYTES, stream>>>(z, cb, wsq, idx);
    vq_gather_kernel<<<BATCH / 8, 256, 0, stream>>>(z, cb, idx, out, partial);
    vq_final_kernel <<<1, 1, 0, stream>>>(partial, out + (size_t)BATCH * DIM);
}